// DecoderBlock_36653250904785
// MI455X (gfx1250) — compile-verified
//
#include <hip/hip_runtime.h>

// Problem constants (from setup_inputs): B=4, M=2048, N=16384, C=D=64
#define BSZ 4
#define MSZ 2048
#define NSZ 16384
#define CH   64      // feat channels (C == D == 64)
#define XCH  128     // concatenated channels (din)
#define OCH  64      // dout

typedef float v2f __attribute__((ext_vector_type(2)));
typedef float v8f __attribute__((ext_vector_type(8)));

// ---------------------------------------------------------------------------
// Kernel 1: inverse-distance-weighted interpolation via V_WMMA_F32_16X16X4_F32
//   For each (b, n): numer[n,d] = sum_m r(n,m) * F[m,d],  denom[n] = sum_m r
//   writes x_cat = concat(feat_up, numer/denom) into workspace (B*N, 128) f32
// ---------------------------------------------------------------------------
#define MCHUNK  128
#define FSTRIDE 72   // 64 + 8 pad: rows k,k+2 differ by 144 dwords == 16 banks

__global__ __launch_bounds__(256) void interp_kernel(
    const float* __restrict__ xyz_down, const float* __restrict__ xyz_up,
    const float* __restrict__ feat_down, const float* __restrict__ feat_up,
    float* __restrict__ xcat)
{
  __shared__ float s_xyz[MCHUNK * 4];          // (x, y, z, |p|^2) per point
  __shared__ float s_feat[MCHUNK * FSTRIDE];

  const int b     = blockIdx.y;
  const int tid   = threadIdx.x;
  const int wave  = tid >> 5;
  const int lane  = tid & 31;
  const int ln    = lane & 15;           // N-index within tile / A-matrix row
  const int khalf = (lane >> 4) << 1;    // 0 for lanes 0-15, 2 for lanes 16-31
  const int n0    = blockIdx.x * 128 + wave * 16;

  // copy feat_up -> xcat[:, 0:64]  (16 rows * 16 float4 per wave, coalesced)
  {
    const float4* fu = (const float4*)feat_up;
    float4*       xc = (float4*)xcat;
#pragma unroll
    for (int j = 0; j < 8; ++j) {
      int  i4  = j * 32 + lane;          // 0..255
      int  row = i4 >> 4;
      int  c4  = i4 & 15;
      long gr  = (long)(b * NSZ + n0 + row);
      xc[gr * 32 + c4] = fu[gr * 16 + c4];
    }
  }

  // this lane's query point (lanes L and L+16 share row ln)
  const float* up = xyz_up + ((long)(b * NSZ + n0 + ln)) * 3;
  const float ux = up[0], uy = up[1], uz = up[2];
  const float u2 = ux * ux + uy * uy + uz * uz;

  v8f z = {0.f, 0.f, 0.f, 0.f, 0.f, 0.f, 0.f, 0.f};
  v8f acc[4];
#pragma unroll
  for (int t = 0; t < 4; ++t) acc[t] = z;
  float rsum = 0.f;

  for (int cstart = 0; cstart < MSZ; cstart += MCHUNK) {
    // stage xyz_down chunk with precomputed |p|^2 in .w
    if (tid < MCHUNK) {
      const float* p = xyz_down + ((long)(b * MSZ + cstart + tid)) * 3;
      float px = p[0], py = p[1], pz = p[2];
      float4 v = {px, py, pz, px * px + py * py + pz * pz};
      *(float4*)&s_xyz[tid * 4] = v;
    }
    // stage feat chunk: 128 rows x 64 floats, bank-padded
    {
      const float4* fd =
          (const float4*)(feat_down + ((long)(b * MSZ + cstart)) * CH);
#pragma unroll
      for (int j = 0; j < 8; ++j) {
        int i4  = j * 256 + tid;         // 0..2047
        int row = i4 >> 4;               // 16 float4 per source row
        int c4  = i4 & 15;
        float4 v = fd[row * 16 + c4];
        *(float4*)&s_feat[row * FSTRIDE + c4 * 4] = v;
      }
    }
    __syncthreads();

    for (int k = 0; k < MCHUNK; k += 4) {
      const int m = k + khalf;
      // build A (16x4 f32 layout): lane ln holds rows m..m+1 of r(n, m)
      float4 p0 = *(const float4*)&s_xyz[m * 4];
      float4 p1 = *(const float4*)&s_xyz[(m + 1) * 4];
      // dist = |u|^2 + |p|^2 - 2 u.p   (matches reference formulation)
      float dot0 = fmaf(ux, p0.x, fmaf(uy, p0.y, uz * p0.z));
      float dot1 = fmaf(ux, p1.x, fmaf(uy, p1.y, uz * p1.z));
      float d0 = fmaf(-2.f, dot0, u2 + p0.w) + 1e-8f;
      float d1 = fmaf(-2.f, dot1, u2 + p1.w) + 1e-8f;
      v2f a;
      a.x = __builtin_amdgcn_rcpf(d0);   // v_rcp_f32: ~1 ulp, avoids IEEE div
      a.y = __builtin_amdgcn_rcpf(d1);
      rsum += a.x + a.y;
#pragma unroll
      for (int t = 0; t < 4; ++t) {
        // B (4x16 f32 layout): v0 = row m (K=k | k+2), v1 = row m+1
        v2f bm;
        bm.x = s_feat[m * FSTRIDE + t * 16 + ln];
        bm.y = s_feat[(m + 1) * FSTRIDE + t * 16 + ln];
        acc[t] = __builtin_amdgcn_wmma_f32_16x16x4_f32(
            false, a, false, bm, (short)0, acc[t], false, false);
      }
    }
    __syncthreads();
  }

  // normalize: denom[row] = rsum(lane) + rsum(lane^16); broadcast per C/D row
  float denom = rsum + __shfl_xor(rsum, 16, 32);
  float inv   = 1.f / denom;             // once per tile: keep exact division
#pragma unroll
  for (int j = 0; j < 8; ++j) {
    float dj  = __shfl(inv, ((lane & 16) >> 1) + j, 32);  // denom of row j(+8)
    int   row = n0 + j + ((lane & 16) ? 8 : 0);
    long  base = (long)(b * NSZ + row) * XCH + 64;
#pragma unroll
    for (int t = 0; t < 4; ++t)
      xcat[base + t * 16 + ln] = acc[t][j] * dj;
  }
}

// ---------------------------------------------------------------------------
// Kernel 2: pointwise MLP  x@W1^T -> BN1+ReLU -> @W2^T -> BN2
//   4 waves/WG, each wave owns 16 rows; LDS tile reused for x then h
// ---------------------------------------------------------------------------
#define K2WAVES 4
#define XSTR    132  // 128 + 4 pad: conflict-free b64 A-reads

__global__ __launch_bounds__(128) void mlp_kernel(
    const float* __restrict__ xcat,
    const float* __restrict__ W1, const float* __restrict__ b1,
    const float* __restrict__ g1, const float* __restrict__ be1,
    const float* __restrict__ mu1, const float* __restrict__ va1,
    const float* __restrict__ W2, const float* __restrict__ b2,
    const float* __restrict__ g2, const float* __restrict__ be2,
    const float* __restrict__ mu2, const float* __restrict__ va2,
    float* __restrict__ out)
{
  __shared__ float s_tile[K2WAVES * 16 * XSTR];

  const int tid     = threadIdx.x;
  const int wave    = tid >> 5;
  const int lane    = tid & 31;
  const int ln      = lane & 15;
  const int khalf   = (lane >> 4) << 1;
  const int rowbase = blockIdx.x * (K2WAVES * 16) + wave * 16;
  float*    tile    = s_tile + wave * 16 * XSTR;

  // fold BN + bias: y = v*s + t  (channel is lane-constant in C/D layout)
  float s1[8], t1[8], s2[4], t2[4];
#pragma unroll
  for (int t = 0; t < 8; ++t) {
    int   c  = t * 16 + ln;
    float sc = g1[c] * rsqrtf(va1[c] + 1e-5f);
    s1[t] = sc;
    t1[t] = (b1[c] - mu1[c]) * sc + be1[c];
  }
#pragma unroll
  for (int t = 0; t < 4; ++t) {
    int   c  = t * 16 + ln;
    float sc = g2[c] * rsqrtf(va2[c] + 1e-5f);
    s2[t] = sc;
    t2[t] = (b2[c] - mu2[c]) * sc + be2[c];
  }

  // stage this wave's 16x128 x-tile (coalesced float4)
  {
    const float4* xc = (const float4*)(xcat + (long)rowbase * XCH);
#pragma unroll
    for (int j = 0; j < 16; ++j) {
      int i4  = j * 32 + lane;           // 0..511
      int row = i4 >> 5;                 // 32 float4 per row
      int c4  = i4 & 31;
      float4 v = xc[row * 32 + c4];
      *(float4*)&tile[row * XSTR + c4 * 4] = v;
    }
  }
  __syncthreads();

  v8f z = {0.f, 0.f, 0.f, 0.f, 0.f, 0.f, 0.f, 0.f};
  v8f h[8];
#pragma unroll
  for (int t = 0; t < 8; ++t) h[t] = z;

  // GEMM1: (16x128) @ W1^T(128x128), K-steps of 4
  for (int k = 0; k < XCH; k += 4) {
    v2f a = *(const v2f*)&tile[ln * XSTR + k + khalf];
#pragma unroll
    for (int t = 0; t < 8; ++t) {
      v2f bm = *(const v2f*)(W1 + (t * 16 + ln) * XCH + k + khalf);
      h[t] = __builtin_amdgcn_wmma_f32_16x16x4_f32(
          false, a, false, bm, (short)0, h[t], false, false);
    }
  }
  __syncthreads();

  // BN1 + ReLU, transpose C/D layout -> row-major LDS (A layout for GEMM2)
#pragma unroll
  for (int t = 0; t < 8; ++t) {
#pragma unroll
    for (int j = 0; j < 8; ++j) {
      int   row = j + ((lane & 16) ? 8 : 0);
      float v   = h[t][j] * s1[t] + t1[t];
      tile[row * XSTR + t * 16 + ln] = v > 0.f ? v : 0.f;
    }
  }
  __syncthreads();

  v8f o[4];
#pragma unroll
  for (int t = 0; t < 4; ++t) o[t] = z;

  // GEMM2: (16x128) @ W2^T(128x64)
  for (int k = 0; k < XCH; k += 4) {
    v2f a = *(const v2f*)&tile[ln * XSTR + k + khalf];
#pragma unroll
    for (int t = 0; t < 4; ++t) {
      v2f bm = *(const v2f*)(W2 + (t * 16 + ln) * XCH + k + khalf);
      o[t] = __builtin_amdgcn_wmma_f32_16x16x4_f32(
          false, a, false, bm, (short)0, o[t], false, false);
    }
  }

  // BN2, store
#pragma unroll
  for (int t = 0; t < 4; ++t) {
#pragma unroll
    for (int j = 0; j < 8; ++j) {
      int row = rowbase + j + ((lane & 16) ? 8 : 0);
      out[(long)row * OCH + t * 16 + ln] = o[t][j] * s2[t] + t2[t];
    }
  }
}

// ---------------------------------------------------------------------------
extern "C" void kernel_launch(void* const* d_in, const int* in_sizes, int n_in,
                              void* d_out, int out_size, void* d_ws,
                              size_t ws_size, hipStream_t stream) {
  const float* xyz_down  = (const float*)d_in[0];
  const float* xyz_up    = (const float*)d_in[1];
  const float* feat_down = (const float*)d_in[2];
  const float* feat_up   = (const float*)d_in[3];
  const float* W1  = (const float*)d_in[4];
  const float* b1  = (const float*)d_in[5];
  const float* g1  = (const float*)d_in[6];
  const float* be1 = (const float*)d_in[7];
  const float* mu1 = (const float*)d_in[8];
  const float* va1 = (const float*)d_in[9];
  const float* W2  = (const float*)d_in[10];
  const float* b2  = (const float*)d_in[11];
  const float* g2  = (const float*)d_in[12];
  const float* be2 = (const float*)d_in[13];
  const float* mu2 = (const float*)d_in[14];
  const float* va2 = (const float*)d_in[15];

  // workspace: x_cat (B*N, 128) f32 = 33.5 MB (L2-resident on MI455X, 192MB)
  float* xcat = (float*)d_ws;

  dim3 g1d(NSZ / 128, BSZ);  // 128 x 4 WGs, 8 waves each, 16 rows/wave
  interp_kernel<<<g1d, 256, 0, stream>>>(xyz_down, xyz_up, feat_down, feat_up,
                                         xcat);

  const int rows = BSZ * NSZ;            // 65536
  mlp_kernel<<<rows / (K2WAVES * 16), 128, 0, stream>>>(
      xcat, W1, b1, g1, be1, mu1, va1, W2, b2, g2, be2, mu2, va2,
      (float*)d_out);
}